// WaveRNN_910533066952
// MI455X (gfx1250) — compile-verified
//
#include <hip/hip_runtime.h>
#include <stdint.h>

// Problem constants (from reference)
#define NSHOTS 4
#define NT     512
#define NZ     256
#define NX     256
#define NRECS  128
#define DT_F   0.001f
#define DH_F   10.0f

// Padded LDS plane:
//  - 1 halo row top/bottom, interior columns live at [4, 259], left halo at col 3,
//    right halo at col 260. Row stride 268 DWORDs: multiple of 4 (16B-aligned b128
//    accesses) and ≡ 12 (mod 64) so a wave of 32 consecutive rows lands on 16
//    distinct bank-quads -> worst case 2-way LDS bank conflict.
#define PW 268
#define PH (NZ + 2)
#define PADWORDS (PW * PH)           // 69144 words
#define LDS_BYTES (PADWORDS * 4)     // 276576 B  (<= 320KB WGP LDS)

// ---------------------------------------------------------------------------
// Init kernel: c2dt2[z,x] = (vp*DT)^2 * (1/DH^2)  -> d_ws scratch (L2-resident,
// streamed every step instead of burning 64 VGPRs/thread on it).
// ---------------------------------------------------------------------------
__global__ __launch_bounds__(256) void c2dt2_init(const float* __restrict__ vp,
                                                  float* __restrict__ c2) {
  int i = blockIdx.x * 256 + threadIdx.x;
  if (i < NZ * NX) {
    float v = vp[i] * DT_F;
    c2[i] = (v * v) * (1.0f / (DH_F * DH_F));   // constant-folded reciprocal
  }
}

// ---------------------------------------------------------------------------
// Persistent wave kernel: one workgroup per shot, whole pressure plane in LDS,
// q = p - p_prev held in VGPRs (64 cells/thread), 512 time steps on-chip.
//   p_{t+1} = p_t + q_t + c2dt2*lap(p_t) (+ src)   ;   q_{t+1} = p_{t+1} - p_t
// ---------------------------------------------------------------------------
__global__ __launch_bounds__(1024) void wave_persistent(
    const float* __restrict__ xwav,     // (NSHOTS, NT)
    const float* __restrict__ c2dt2,    // (NZ, NX) in d_ws
    const int*   __restrict__ src_z,    // (NSHOTS)
    const int*   __restrict__ src_x,    // (NSHOTS)
    const int*   __restrict__ rec_z,    // (NSHOTS, NRECS)
    const int*   __restrict__ rec_x,    // (NSHOTS, NRECS)
    float*       __restrict__ out)      // (NSHOTS, NT, NRECS, 1)
{
  extern __shared__ float P[];          // padded pressure plane, PADWORDS floats

  const int s   = blockIdx.x;
  const int tid = threadIdx.x;

  // Zero-fill LDS (interior + halo ring). Halo is never written again.
  for (int i = tid; i < PADWORDS; i += 1024) P[i] = 0.0f;

  // Thread -> 64-cell strip. row = tid & 255 puts 32 consecutive rows in one
  // wave (16 distinct bank-quads at stride 268 -> ~2-way conflicts on b128).
  const int row     = tid & 255;        // 0..255
  const int colbase = (tid >> 8) * 64;  // 0,64,128,192

  float* up = &P[(row    ) * PW + 4 + colbase];
  float* ce = &P[(row + 1) * PW + 4 + colbase];
  float* dn = &P[(row + 2) * PW + 4 + colbase];
  const float* c2row = c2dt2 + row * NX + colbase;

  // Source ownership + software-pipelined wavelet value.
  const int  sz      = src_z[s];
  const int  sx      = src_x[s];
  const bool own_src = (row == sz) && (sx >= colbase) && (sx < colbase + 64);
  const int  src_off = sx - colbase;
  const float* xw    = xwav + s * NT;
  float w = own_src ? xw[0] : 0.0f;     // wavelet for step t, loaded one step early

  // Receiver setup (threads 0..127): LDS byte offset + global element pointer.
  unsigned     rec_lds = 0;
  const float* rec_ptr = out;
  if (tid < NRECS) {
    int rz = rec_z[s * NRECS + tid];
    int rx = rec_x[s * NRECS + tid];
    rec_lds = (unsigned)((((rz + 1) * PW) + 4 + rx) * 4);
    rec_ptr = out + (size_t)s * NT * NRECS + tid;
  }

  // q = p - p_prev, per-cell, in VGPRs.
  float q[64];
#pragma unroll
  for (int i = 0; i < 64; ++i) q[i] = 0.0f;

  __syncthreads();

  for (int t = 0; t < NT; ++t) {
    // ---- Phase 1: q += c2dt2 * lap(p)   (LDS reads only) ----
    float c_im1 = ce[-1];               // left neighbor of first cell (halo-safe)
#pragma unroll
    for (int g = 0; g < 16; ++g) {
      const float4 pc = *reinterpret_cast<const float4*>(ce + 4 * g);
      const float4 pu = *reinterpret_cast<const float4*>(up + 4 * g);
      const float4 pd = *reinterpret_cast<const float4*>(dn + 4 * g);
      const float4 k  = *reinterpret_cast<const float4*>(c2row + 4 * g);
      const float  c_ip1 = ce[4 * g + 4];   // next group's first / right halo
      q[4 * g + 0] += k.x * (pu.x + pd.x + c_im1 + pc.y - 4.0f * pc.x);
      q[4 * g + 1] += k.y * (pu.y + pd.y + pc.x  + pc.z - 4.0f * pc.y);
      q[4 * g + 2] += k.z * (pu.z + pd.z + pc.y  + pc.w - 4.0f * pc.z);
      q[4 * g + 3] += k.w * (pu.w + pd.w + pc.z  + c_ip1 - 4.0f * pc.w);
      c_im1 = pc.w;
    }

    // Drain previous step's async receiver stores before anyone rewrites p.
    asm volatile("s_wait_asynccnt 0x0" ::: "memory");
    __syncthreads();   // all stencil reads of p complete

    // ---- Phase 2: source inject + p += q (each thread RMWs own cells) ----
    if (own_src) q[src_off] += w * (DT_F * DT_F);
#pragma unroll
    for (int g = 0; g < 16; ++g) {
      float4 v = *reinterpret_cast<const float4*>(ce + 4 * g);
      v.x += q[4 * g + 0];
      v.y += q[4 * g + 1];
      v.z += q[4 * g + 2];
      v.w += q[4 * g + 3];
      *reinterpret_cast<float4*>(ce + 4 * g) = v;
    }
    __syncthreads();   // p fully advanced to p_{t+1}

    // ---- Record receivers: async LDS -> global (CDNA5, ASYNCcnt-tracked).
    // Overlaps next step's phase-1 stencil; drained before next phase-2 writes.
    if (tid < NRECS) {
      uint64_t ga = (uint64_t)(uintptr_t)(rec_ptr + (size_t)t * NRECS);
      asm volatile("global_store_async_from_lds_b32 %0, %1, off"
                   :: "v"(ga), "v"(rec_lds)
                   : "memory");
    }

    // Prefetch next step's wavelet; retires during next phase-1 stencil.
    if (own_src && (t + 1 < NT)) w = xw[t + 1];
  }

  asm volatile("s_wait_asynccnt 0x0" ::: "memory");
}

// ---------------------------------------------------------------------------
// Host launcher
// Inputs: 0=x (NSHOTS,NT) f32, 1=vp (NZ,NX) f32, 2=src_z i32, 3=src_x i32,
//         4=rec_z (NSHOTS,NRECS) i32, 5=rec_x (NSHOTS,NRECS) i32
// Output: (NSHOTS, NT, NRECS, 1) f32
// ---------------------------------------------------------------------------
extern "C" void kernel_launch(void* const* d_in, const int* in_sizes, int n_in,
                              void* d_out, int out_size, void* d_ws, size_t ws_size,
                              hipStream_t stream) {
  const float* x     = (const float*)d_in[0];
  const float* vp    = (const float*)d_in[1];
  const int*   src_z = (const int*)d_in[2];
  const int*   src_x = (const int*)d_in[3];
  const int*   rec_z = (const int*)d_in[4];
  const int*   rec_x = (const int*)d_in[5];
  float*       out   = (float*)d_out;
  float*       c2    = (float*)d_ws;       // NZ*NX floats of scratch

  // Allow >64KB dynamic LDS (276KB plane needs the CDNA5 320KB WGP LDS).
  (void)hipFuncSetAttribute(reinterpret_cast<const void*>(wave_persistent),
                            hipFuncAttributeMaxDynamicSharedMemorySize,
                            LDS_BYTES);

  c2dt2_init<<<(NZ * NX + 255) / 256, 256, 0, stream>>>(vp, c2);
  wave_persistent<<<NSHOTS, 1024, LDS_BYTES, stream>>>(
      x, c2, src_z, src_x, rec_z, rec_x, out);
}